// PointPillar_9122510536946
// MI455X (gfx1250) — compile-verified
//
#include <hip/hip_runtime.h>
#include <stdint.h>

// ---------------------------------------------------------------------------
// PointPillars head post-processing for MI455X (gfx1250, wave32).
// Bandwidth-bound: no matmul content, so no WMMA; the CDNA5 win is the
// Tensor Data Mover streaming 36KB class-score tiles into LDS per block.
// ---------------------------------------------------------------------------

#define NUMC   3
#define CODE   7
#define NMSPRE 1000
#define HH     512
#define WW     512
#define AA     6
#define HW     (HH * WW)        // 262144
#define NANCH  (HW * AA)        // 1572864
#define EQ_CAP 4096

// workspace layout in uint32 words (total ~6.32 MB)
#define WS_KEYS   0
#define WS_HIST   (NANCH)
#define WS_CTRL   (NANCH + 256)            // [0]=prefix [1]=kneed [2]=cnt_gt [3]=cnt_eq
#define WS_SELKEY (NANCH + 256 + 4)
#define WS_SELIDX (NANCH + 256 + 4 + 1024)
#define WS_EQ     (NANCH + 256 + 4 + 2048)

typedef unsigned int uint32x4 __attribute__((ext_vector_type(4)));
typedef int          int32x8  __attribute__((ext_vector_type(8)));
typedef int          int32x4  __attribute__((ext_vector_type(4)));

#ifndef __has_builtin
#define __has_builtin(x) 0
#endif
#if __has_builtin(__builtin_amdgcn_tensor_load_to_lds)
#define USE_TDM 1
#else
#define USE_TDM 0
#endif

// float -> order-preserving sortable uint32 (ascending)
__device__ __forceinline__ uint32_t fkey(float f) {
  uint32_t u = __float_as_uint(f);
  return (u & 0x80000000u) ? ~u : (u | 0x80000000u);
}

// raw LDS byte offset of a __shared__ object (generic -> addrspace(3) -> int)
__device__ __forceinline__ uint32_t lds_off(const void* p) {
  return (uint32_t)(uint64_t)(__attribute__((address_space(3))) const char*)p;
}

// ---------------------------------------------------------------------------
// Phase 1: per-anchor max class logit -> sortable key.
// One block per image row h. TDM loads a 2-D tile: tile_dim0 = 512 (contiguous
// x within a plane), tile_dim1 = 18 (planes), tensor_dim0_stride = H*W.
// LDS result: tile[plane*512 + w], conflict-free reads (consecutive lanes ->
// consecutive banks).
// ---------------------------------------------------------------------------
__global__ __launch_bounds__(WW) void k_phase1(const float* __restrict__ cls,
                                               uint32_t* __restrict__ keys) {
  __shared__ float    tile[AA * NUMC * WW];   // 18*512 floats = 36 KB
  __shared__ uint32_t klds[WW * AA];          // 3072 keys     = 12 KB
  const int row = blockIdx.x;
  const int tid = threadIdx.x;

#if USE_TDM
  if (tid == 0) {
    uint64_t gaddr = (uint64_t)(uintptr_t)(cls + (size_t)row * WW);
    uint32_t laddr = lds_off(&tile[0]);
    uint32x4 g0;
    g0[0] = 1u;                                   // count=1, user descriptor
    g0[1] = laddr;                                // lds_addr
    g0[2] = (uint32_t)gaddr;                      // global_addr[31:0]
    g0[3] = ((uint32_t)(gaddr >> 32) & 0x01FFFFFFu) | (2u << 30); // addr[56:32], type=2
    int32x8 g1;
    g1[0] = (int)(2u << 16);                      // data_size = 2 (4 bytes)
    g1[1] = (int)((uint32_t)WW << 16);            // tensor_dim0[15:0] = 512
    g1[2] = (int)((uint32_t)(AA * NUMC) << 16);   // dim0 hi=0 | tensor_dim1 = 18
    g1[3] = (int)((uint32_t)WW << 16);            // dim1 hi=0 | tile_dim0 = 512
    g1[4] = AA * NUMC;                            // tile_dim1 = 18, tile_dim2 = 0
    g1[5] = HW;                                   // tensor_dim0_stride[31:0] = 262144
    g1[6] = 0;                                    // stride hi / dim1_stride lo
    g1[7] = 0;                                    // dim1_stride hi (unused, 2-D)
    int32x4 g2 = {0, 0, 0, 0};                    // group 2 (unused, 2-D tensor)
    int32x4 g3 = {0, 0, 0, 0};                    // group 3 (unused, 2-D tensor)
    int32x8 g4 = {0, 0, 0, 0, 0, 0, 0, 0};        // extra group (clang-23 6-arg form)
    __builtin_amdgcn_tensor_load_to_lds(g0, g1, g2, g3, g4, 0);
#if __has_builtin(__builtin_amdgcn_s_wait_tensorcnt)
    __builtin_amdgcn_s_wait_tensorcnt(0);
#else
    asm volatile("s_wait_tensorcnt 0x0" ::: "memory");
#endif
  }
  __syncthreads();
#else
  #pragma unroll
  for (int p = 0; p < AA * NUMC; ++p)
    tile[p * WW + tid] = cls[(size_t)p * HW + (size_t)row * WW + tid];
  __syncthreads();
#endif

  const int w = tid;
  #pragma unroll
  for (int a = 0; a < AA; ++a) {
    float m = tile[(a * 3 + 0) * WW + w];
    m = fmaxf(m, tile[(a * 3 + 1) * WW + w]);
    m = fmaxf(m, tile[(a * 3 + 2) * WW + w]);
    klds[w * AA + a] = fkey(m);   // sigmoid monotone: order == order of logits
  }
  __syncthreads();
  uint32_t* dst = keys + (size_t)row * (WW * AA);
  for (int t = tid; t < WW * AA; t += WW) dst[t] = klds[t];  // coalesced
}

// ---------------------------------------------------------------------------
// Radix-select (MSB-first, 8 bits/pass) for the exact 1000th-largest key.
// ---------------------------------------------------------------------------
__global__ void k_init(uint32_t* ws) {
  int i = threadIdx.x;
  if (i < 256) ws[WS_HIST + i] = 0u;
  if (i == 0) {
    ws[WS_CTRL + 0] = 0u;       // prefix
    ws[WS_CTRL + 1] = NMSPRE;   // k remaining
    ws[WS_CTRL + 2] = 0u;       // cnt_gt
    ws[WS_CTRL + 3] = 0u;       // cnt_eq
  }
}

__global__ __launch_bounds__(256) void k_hist(const uint32_t* __restrict__ keys,
                                              uint32_t* __restrict__ ws, int shift) {
  __shared__ uint32_t lh[256];
  const int t = threadIdx.x;
  lh[t] = 0u;
  __syncthreads();
  const uint32_t prefix = ws[WS_CTRL + 0];
  const uint32_t mhi = (shift >= 24) ? 0u : (0xFFFFFFFFu << (shift + 8));
  for (size_t i = (size_t)blockIdx.x * 256 + t; i < NANCH; i += (size_t)gridDim.x * 256) {
    uint32_t k = keys[i];
    if ((k & mhi) == (prefix & mhi)) atomicAdd(&lh[(k >> shift) & 0xFFu], 1u);
  }
  __syncthreads();
  if (lh[t]) atomicAdd(&ws[WS_HIST + t], lh[t]);
}

__global__ void k_scan(uint32_t* ws, int shift) {
  if (threadIdx.x != 0) return;
  uint32_t k = ws[WS_CTRL + 1];
  int d = 255;
  for (; d > 0; --d) {
    uint32_t c = ws[WS_HIST + d];
    if (c >= k) break;
    k -= c;                       // elements strictly above chosen digit
  }
  ws[WS_CTRL + 0] |= ((uint32_t)d) << shift;
  ws[WS_CTRL + 1] = k;
  for (int i = 0; i < 256; ++i) ws[WS_HIST + i] = 0u;  // reset for next pass
}

// key > T -> definitely selected; key == T -> boundary ties
__global__ __launch_bounds__(256) void k_compact(const uint32_t* __restrict__ keys,
                                                 uint32_t* __restrict__ ws) {
  const uint32_t T = ws[WS_CTRL + 0];
  for (size_t i = (size_t)blockIdx.x * 256 + threadIdx.x; i < NANCH;
       i += (size_t)gridDim.x * 256) {
    uint32_t k = keys[i];
    if (k > T) {
      uint32_t p = atomicAdd(&ws[WS_CTRL + 2], 1u);
      if (p < NMSPRE) { ws[WS_SELKEY + p] = k; ws[WS_SELIDX + p] = (uint32_t)i; }
    } else if (k == T) {
      uint32_t q = atomicAdd(&ws[WS_CTRL + 3], 1u);
      if (q < EQ_CAP) ws[WS_EQ + q] = (uint32_t)i;
    }
  }
}

// JAX tie-break: among equal values, keep smallest indices (rank by index).
__global__ __launch_bounds__(1024) void k_ties(uint32_t* ws) {
  const uint32_t T    = ws[WS_CTRL + 0];
  const uint32_t need = ws[WS_CTRL + 1];
  const uint32_t base = ws[WS_CTRL + 2];       // == NMSPRE - need
  uint32_t cnt = ws[WS_CTRL + 3];
  if (cnt > EQ_CAP) cnt = EQ_CAP;
  for (uint32_t i = threadIdx.x; i < cnt; i += 1024) {
    uint32_t v = ws[WS_EQ + i];
    uint32_t r = 0;
    for (uint32_t j = 0; j < cnt; ++j) r += (ws[WS_EQ + j] < v) ? 1u : 0u;
    if (r < need && base + r < NMSPRE) {
      ws[WS_SELKEY + base + r] = T;
      ws[WS_SELIDX + base + r] = v;
    }
  }
}

// Single-block bitonic sort of 1024 (1000 + pad) by (key desc, index asc).
// Makes output deterministic despite atomic-append ordering.
__global__ __launch_bounds__(1024) void k_sort(uint32_t* ws) {
  __shared__ uint32_t sk[1024], si[1024];
  const int i = threadIdx.x;
  if (i < NMSPRE) { sk[i] = ws[WS_SELKEY + i]; si[i] = ws[WS_SELIDX + i]; }
  else            { sk[i] = 0u;                si[i] = 0xFFFFFFFFu;       }
  __syncthreads();
  for (int k = 2; k <= 1024; k <<= 1) {
    for (int j = k >> 1; j > 0; j >>= 1) {
      int ixj = i ^ j;
      if (ixj > i) {
        uint32_t ka = sk[i], ia = si[i], kb = sk[ixj], ib = si[ixj];
        bool before = (ka > kb) || (ka == kb && ia < ib);  // i-elem precedes ixj-elem
        bool asc = ((i & k) == 0);
        if (asc ? !before : before) {
          sk[i] = kb; si[i] = ib; sk[ixj] = ka; si[ixj] = ia;
        }
      }
      __syncthreads();
    }
  }
  if (i < NMSPRE) { ws[WS_SELKEY + i] = sk[i]; ws[WS_SELIDX + i] = si[i]; }
}

// ---------------------------------------------------------------------------
// Gather + decode for the 1000 winners only.
// n = (h*W + w)*A + a ; plane value at channel ch lives at ch*HW + (h*W + w).
// ---------------------------------------------------------------------------
__global__ __launch_bounds__(256) void k_decode(const float* __restrict__ cls,
                                                const float* __restrict__ bb,
                                                const float* __restrict__ dir,
                                                const float* __restrict__ an,
                                                const uint32_t* __restrict__ ws,
                                                float* __restrict__ out) {
  int i = blockIdx.x * blockDim.x + threadIdx.x;
  if (i >= NMSPRE) return;
  uint32_t n = ws[WS_SELIDX + i];
  int a = (int)(n % AA);
  size_t pos = (size_t)(n / AA);

  // scores = sigmoid(logits)
  #pragma unroll
  for (int c = 0; c < NUMC; ++c) {
    float l = cls[(size_t)(a * NUMC + c) * HW + pos];
    out[7 * NMSPRE + i * NUMC + c] = 1.0f / (1.0f + expf(-l));
  }

  float t[CODE], av[CODE];
  #pragma unroll
  for (int j = 0; j < CODE; ++j) t[j] = bb[(size_t)(a * CODE + j) * HW + pos];
  #pragma unroll
  for (int j = 0; j < CODE; ++j) av[j] = an[(size_t)n * CODE + j];

  float xa = av[0], ya = av[1], za = av[2], wa = av[3], la = av[4], ha = av[5], ra = av[6];
  za += 0.5f * ha;
  float diag = sqrtf(la * la + wa * wa);
  float xg = t[0] * diag + xa;
  float yg = t[1] * diag + ya;
  float zg = t[2] * ha + za;
  float wg = expf(t[3]) * wa;
  float lg = expf(t[4]) * la;
  float hg = expf(t[5]) * ha;
  float rg = t[6] + ra;
  zg -= 0.5f * hg;
  out[i * 7 + 0] = xg; out[i * 7 + 1] = yg; out[i * 7 + 2] = zg;
  out[i * 7 + 3] = wg; out[i * 7 + 4] = lg; out[i * 7 + 5] = hg; out[i * 7 + 6] = rg;

  float d0 = dir[(size_t)(a * 2 + 0) * HW + pos];
  float d1 = dir[(size_t)(a * 2 + 1) * HW + pos];
  out[10 * NMSPRE + i] = (d1 > d0) ? 1.0f : 0.0f;   // argmax, ties -> 0
}

// ---------------------------------------------------------------------------
extern "C" void kernel_launch(void* const* d_in, const int* in_sizes, int n_in,
                              void* d_out, int out_size, void* d_ws, size_t ws_size,
                              hipStream_t stream) {
  const float* cls = (const float*)d_in[0];   // (18, 512, 512)
  const float* bb  = (const float*)d_in[1];   // (42, 512, 512)
  const float* dir = (const float*)d_in[2];   // (12, 512, 512)
  const float* an  = (const float*)d_in[3];   // (N, 7)
  uint32_t* ws  = (uint32_t*)d_ws;            // needs ~6.4 MB
  float*    out = (float*)d_out;              // 7000 + 3000 + 1000

  k_init  <<<1, 256, 0, stream>>>(ws);
  k_phase1<<<HH, WW, 0, stream>>>(cls, ws + WS_KEYS);
  for (int pass = 0; pass < 4; ++pass) {
    int shift = 24 - 8 * pass;
    k_hist<<<1536, 256, 0, stream>>>(ws + WS_KEYS, ws, shift);
    k_scan<<<1, 1, 0, stream>>>(ws, shift);
  }
  k_compact<<<1536, 256, 0, stream>>>(ws + WS_KEYS, ws);
  k_ties  <<<1, 1024, 0, stream>>>(ws);
  k_sort  <<<1, 1024, 0, stream>>>(ws);
  k_decode<<<(NMSPRE + 255) / 256, 256, 0, stream>>>(cls, bb, dir, an, ws, out);
}